// ChebConvLayer_53532472377790
// MI455X (gfx1250) — compile-verified
//
#include <hip/hip_runtime.h>

#define VNODES 100000
#define NEDGES 1600000
#define NB 4
#define FIN 64
#define FOUT 64
#define ROW 256          // NB * FIN
#define ROWELEMS ((size_t)VNODES * ROW)

typedef float v2f __attribute__((ext_vector_type(2)));
typedef float v8f __attribute__((ext_vector_type(8)));

// x [B,V,FIN] -> x0 [V, B*FIN]   (x0[v, b*64+i] = x[b,v,i])
__global__ __launch_bounds__(256) void make_x0_kernel(const float* __restrict__ x,
                                                      float* __restrict__ x0) {
    int v = blockIdx.x;
    int t = threadIdx.x;          // 0..255 : b = t>>6, i = t&63
    int b = t >> 6;
    int i = t & 63;
    x0[(size_t)v * ROW + t] = x[((size_t)b * VNODES + v) * FIN + i];
}

__global__ __launch_bounds__(256) void zero_kernel(float* __restrict__ dst) {
    size_t idx = (size_t)blockIdx.x * 256 + threadIdx.x;
    dst[idx] = 0.0f;
}

// dst = -src   (seed for the Chebyshev recurrence  x_k = 2 L x_{k-1} - x_{k-2})
__global__ __launch_bounds__(256) void negcopy_kernel(float* __restrict__ dst,
                                                      const float* __restrict__ src) {
    size_t idx = (size_t)blockIdx.x * 256 + threadIdx.x;
    dst[idx] = -src[idx];
}

// dst[rows[e], :] += scale * vals[e] * src[cols[e], :]
// one wave (32 lanes) per edge; 8 floats (2 x float4, at float4 index lane and lane+32) per lane.
// Both matrices are 102.4 MB -> L2-resident; atomics resolve in L2.
__global__ __launch_bounds__(256) void spmm_scatter_kernel(const float* __restrict__ src,
                                                           float* __restrict__ dst,
                                                           const float* __restrict__ vals,
                                                           const int* __restrict__ rows,
                                                           const int* __restrict__ cols,
                                                           float scale) {
    int wid  = threadIdx.x >> 5;                // 8 waves / block
    int lane = threadIdx.x & 31;
    int e = blockIdx.x * 8 + wid;
    if (e >= NEDGES) return;

    int   r = rows[e];
    int   c = cols[e];
    float v = vals[e] * scale;

    const float4* s = (const float4*)(src + (size_t)c * ROW);
    float*        d = dst + (size_t)r * ROW;

#pragma unroll
    for (int half = 0; half < 2; ++half) {
        int qi = lane + half * 32;              // float4 index 0..63
        float4 f = s[qi];
        int base = qi * 4;
        unsafeAtomicAdd(d + base + 0, v * f.x);
        unsafeAtomicAdd(d + base + 1, v * f.y);
        unsafeAtomicAdd(d + base + 2, v * f.z);
        unsafeAtomicAdd(d + base + 3, v * f.w);
    }
}

// out[b, v0:v0+16, o0:o0+16] (+)= T_k[v0:v0+16, b*64 : b*64+64] @ Wk[64, o0:o0+16]
// using V_WMMA_F32_16X16X4_F32, 16 steps over the i=64 reduction.
// grid.x = (V/16)*NB, 4 waves per block = the 4 o-tiles (share A rows in cache).
__global__ __launch_bounds__(128) void cheb_gemm_wmma_kernel(const float* __restrict__ T,
                                                             const float* __restrict__ Wk,
                                                             const float* __restrict__ bias,
                                                             float* __restrict__ out,
                                                             int initWithBias) {
    int lane  = threadIdx.x & 31;
    int otile = threadIdx.x >> 5;               // 0..3
    int b     = blockIdx.x & (NB - 1);          // NB == 4
    int vtile = blockIdx.x >> 2;

    int v0 = vtile * 16;
    int o0 = otile * 16;
    int n    = lane & 15;                       // N (and M) index within tile
    int hi   = lane >> 4;                       // 0: K 0,1 / rows 0..7 ; 1: K 2,3 / rows 8..15
    int koff = hi * 2;
    int o    = o0 + n;

    v8f c;
    if (initWithBias) {
        float bv = bias[o];
#pragma unroll
        for (int r = 0; r < 8; ++r) c[r] = bv;
    } else {
#pragma unroll
        for (int r = 0; r < 8; ++r) {
            int vr = v0 + r + hi * 8;
            c[r] = out[((size_t)b * VNODES + vr) * FOUT + o];
        }
    }

    // A fragment base: row m = v0 + (lane&15), columns b*64 + kk + koff (+0,+1)
    const float* Arow = T + (size_t)(v0 + n) * ROW + b * FIN + koff;

#pragma unroll
    for (int kk = 0; kk < FIN; kk += 4) {
        v2f a = *(const v2f*)(Arow + kk);                       // 8B aligned
        v2f bb;
        bb.x = Wk[(kk + koff)     * FOUT + o];                  // B row K=kk+koff
        bb.y = Wk[(kk + koff + 1) * FOUT + o];                  // B row K=kk+koff+1
        c = __builtin_amdgcn_wmma_f32_16x16x4_f32(false, a, false, bb,
                                                  (short)0, c, false, false);
    }

#pragma unroll
    for (int r = 0; r < 8; ++r) {
        int vr = v0 + r + hi * 8;
        out[((size_t)b * VNODES + vr) * FOUT + o] = c[r];
    }
}

extern "C" void kernel_launch(void* const* d_in, const int* in_sizes, int n_in,
                              void* d_out, int out_size, void* d_ws, size_t ws_size,
                              hipStream_t stream) {
    const float* x      = (const float*)d_in[0];   // [B,V,FIN]
    const float* weight = (const float*)d_in[1];   // [K,FIN,FOUT]
    const float* bias   = (const float*)d_in[2];   // [FOUT]
    const float* vals   = (const float*)d_in[3];   // [E]
    const int*   rows   = (const int*)d_in[4];     // [E]
    const int*   cols   = (const int*)d_in[5];     // [E]
    float*       out    = (float*)d_out;           // [B,V,FOUT]

    float* b0 = (float*)d_ws;                      // rotating [V, 256] buffers
    float* b1 = b0 + ROWELEMS;
    float* b2 = b1 + ROWELEMS;

    const int gElem  = (int)(ROWELEMS / 256);      // 100000 blocks for elementwise
    const int gEdge  = NEDGES / 8;                 // 200000 blocks, 8 edges each
    const int gGemm  = (VNODES / 16) * NB;         // 25000 blocks, 4 waves each

    // T0 = x0
    make_x0_kernel<<<VNODES, 256, 0, stream>>>(x, b0);
    cheb_gemm_wmma_kernel<<<gGemm, 128, 0, stream>>>(b0, weight + 0 * FIN * FOUT, bias, out, 1);

    // T1 = L @ T0
    zero_kernel<<<gElem, 256, 0, stream>>>(b1);
    spmm_scatter_kernel<<<gEdge, 256, 0, stream>>>(b0, b1, vals, rows, cols, 1.0f);
    cheb_gemm_wmma_kernel<<<gGemm, 128, 0, stream>>>(b1, weight + 1 * FIN * FOUT, bias, out, 0);

    // T2 = 2 L @ T1 - T0
    negcopy_kernel<<<gElem, 256, 0, stream>>>(b2, b0);
    spmm_scatter_kernel<<<gEdge, 256, 0, stream>>>(b1, b2, vals, rows, cols, 2.0f);
    cheb_gemm_wmma_kernel<<<gGemm, 128, 0, stream>>>(b2, weight + 2 * FIN * FOUT, bias, out, 0);

    // T3 = 2 L @ T2 - T1   (reuses b0)
    negcopy_kernel<<<gElem, 256, 0, stream>>>(b0, b1);
    spmm_scatter_kernel<<<gEdge, 256, 0, stream>>>(b2, b0, vals, rows, cols, 2.0f);
    cheb_gemm_wmma_kernel<<<gGemm, 128, 0, stream>>>(b0, weight + 3 * FIN * FOUT, bias, out, 0);
}